// DependencyTreeRelModel_21552145891578
// MI455X (gfx1250) — compile-verified
//
#include <hip/hip_runtime.h>
#include <math.h>

// ---------------- problem constants (match reference) ----------------
#define NN    3072
#define HH    256
#define NRELC 18
#define EE    (NN - 1)          // 3071 edges
#define EHALF (EE / 2)          // 1535 : left/right split
#define EPAD  NN                // padded edge rows for WMMA GEMM
#define LUB   64                // LU panel width
#define ALPHA 0.25f

typedef float v2f __attribute__((ext_vector_type(2)));
typedef float v8f __attribute__((ext_vector_type(8)));

__device__ __forceinline__ float gelu_exact(float v) {
    return 0.5f * v * (1.0f + erff(v * 0.7071067811865476f));
}

// fp32 WMMA wrapper: D = A(16x4) @ B(4x16) + C
__device__ __forceinline__ v8f wmma4(v2f a, v2f b, v8f c) {
    return __builtin_amdgcn_wmma_f32_16x16x4_f32(false, a, false, b,
                                                 (short)0, c, false, false);
}

// ---------------------------------------------------------------------
// Single 16x16 tile, A given TRANSPOSED (AT is [K x M], ldat = M stride)
// so A-fragment loads are lane-contiguous; B row-major [K x N].
// A frag: lane = m + 16*(k>=2), reg = k&1  -> AT[k][m0+l16]
// B frag: lane = n + 16*(k>=2), reg = k&1  -> B[k][n0+l16]
// C/D: reg r holds m = r + 8*(lane>=16), n = lane&15.
// ---------------------------------------------------------------------
__device__ __forceinline__ v8f wmma_tile_AT(const float* __restrict__ AT,
                                            const float* __restrict__ B,
                                            int ldat, int ldb, int K,
                                            int row0, int col0) {
    const int lane = threadIdx.x & 31;
    const int half = lane >> 4;
    const int l16  = lane & 15;
    v8f acc = {};
    for (int k = 0; k < K; k += 4) {
        const int ka = k + half * 2;
        v2f a, b;
        a.x = AT[(size_t)(ka + 0) * ldat + row0 + l16];
        a.y = AT[(size_t)(ka + 1) * ldat + row0 + l16];
        b.x = B[(size_t)(ka + 0) * ldb + col0 + l16];
        b.y = B[(size_t)(ka + 1) * ldb + col0 + l16];
        acc = wmma4(a, b, acc);
    }
    return acc;
}

// ---------------- tiled transpose: out[K x M] = in[M x K]^T ----------
__global__ void transpose_kernel(const float* __restrict__ in,
                                 float* __restrict__ out, int M, int K) {
    __shared__ float tile[32][33];
    const int bx = blockIdx.x * 32;   // K offset
    const int by = blockIdx.y * 32;   // M offset
    const int tx = threadIdx.x, ty = threadIdx.y;   // 32 x 8
    #pragma unroll
    for (int r = 0; r < 32; r += 8)
        tile[ty + r][tx] = in[(size_t)(by + ty + r) * K + bx + tx];
    __syncthreads();
    #pragma unroll
    for (int r = 0; r < 32; r += 8)
        out[(size_t)(bx + ty + r) * M + by + tx] = tile[tx][ty + r];
}

// ---------------- GEMM (AT layout), dual store C and C^T -------------
__global__ void gemm_at_dual_kernel(const float* __restrict__ AT,
                                    const float* __restrict__ B,
                                    float* __restrict__ C,
                                    float* __restrict__ CT,
                                    int M, int Ncols, int K, int ldat) {
    const int tile  = blockIdx.x * blockDim.y + threadIdx.y;
    const int tcols = Ncols >> 4;
    const int bm = tile / tcols, bn = tile % tcols;
    if (bm >= (M >> 4)) return;
    v8f acc = wmma_tile_AT(AT, B, ldat, Ncols, K, bm * 16, bn * 16);
    const int lane = threadIdx.x & 31, half = lane >> 4, l16 = lane & 15;
    const int n = bn * 16 + l16;
    #pragma unroll
    for (int r = 0; r < 8; ++r) {
        const int m = bm * 16 + r + 8 * half;
        C[(size_t)m * Ncols + n] = acc[r];
        CT[(size_t)n * M + m]    = acc[r];
    }
}

// ---------------- GEMM (AT layout) + bias + exact GELU ---------------
__global__ void gemm_at_bias_gelu_kernel(const float* __restrict__ AT,
                                         const float* __restrict__ B,
                                         const float* __restrict__ bias,
                                         float* __restrict__ C,
                                         int M, int Ncols, int K, int ldat) {
    const int tile  = blockIdx.x * blockDim.y + threadIdx.y;
    const int tcols = Ncols >> 4;
    const int bm = tile / tcols, bn = tile % tcols;
    if (bm >= (M >> 4)) return;
    v8f acc = wmma_tile_AT(AT, B, ldat, Ncols, K, bm * 16, bn * 16);
    const int lane = threadIdx.x & 31, half = lane >> 4, l16 = lane & 15;
    const float bv = bias[bn * 16 + l16];
    #pragma unroll
    for (int r = 0; r < 8; ++r) {
        const int m = bm * 16 + r + 8 * half;
        C[(size_t)m * Ncols + bn * 16 + l16] = gelu_exact(acc[r] + bv);
    }
}

// ---------------- A = exp(compat0)+exp(compat1)  [N x N] -------------
// 32x32 output per wave (2x2 tiles), dual product sharing B fragments.
// All operands pre-transposed -> every fragment load is coalesced.
__global__ void compat_exp_kernel(const float* __restrict__ xW0T,  // [H x N]
                                  const float* __restrict__ xW1T,  // [H x N]
                                  const float* __restrict__ xT,    // [H x N]
                                  const float* __restrict__ hd,    // [N][4]
                                  const float* __restrict__ b_bilin,
                                  float* __restrict__ Amat) {
    const int wave = blockIdx.x * blockDim.y + threadIdx.y;
    const int tc   = NN / 32;                       // 96
    const int bm = wave / tc, bn = wave % tc;
    if (bm >= tc) return;
    const int lane = threadIdx.x & 31, half = lane >> 4, l16 = lane & 15;
    const int i0 = bm * 32, j0 = bn * 32;

    v8f c0[2][2] = {};
    v8f c1[2][2] = {};
    for (int k = 0; k < HH; k += 4) {
        const int ka = k + half * 2;
        const float* A0a = xW0T + (size_t)ka * NN;
        const float* A0b = xW0T + (size_t)(ka + 1) * NN;
        const float* A1a = xW1T + (size_t)ka * NN;
        const float* A1b = xW1T + (size_t)(ka + 1) * NN;
        const float* Ba  = xT   + (size_t)ka * NN;
        const float* Bb  = xT   + (size_t)(ka + 1) * NN;
        v2f a00, a01, a10, a11, b0, b1;
        a00.x = A0a[i0 + l16];       a00.y = A0b[i0 + l16];
        a01.x = A0a[i0 + 16 + l16];  a01.y = A0b[i0 + 16 + l16];
        a10.x = A1a[i0 + l16];       a10.y = A1b[i0 + l16];
        a11.x = A1a[i0 + 16 + l16];  a11.y = A1b[i0 + 16 + l16];
        b0.x  = Ba[j0 + l16];        b0.y  = Bb[j0 + l16];
        b1.x  = Ba[j0 + 16 + l16];   b1.y  = Bb[j0 + 16 + l16];
        c0[0][0] = wmma4(a00, b0, c0[0][0]);
        c0[0][1] = wmma4(a00, b1, c0[0][1]);
        c0[1][0] = wmma4(a01, b0, c0[1][0]);
        c0[1][1] = wmma4(a01, b1, c0[1][1]);
        c1[0][0] = wmma4(a10, b0, c1[0][0]);
        c1[0][1] = wmma4(a10, b1, c1[0][1]);
        c1[1][0] = wmma4(a11, b0, c1[1][0]);
        c1[1][1] = wmma4(a11, b1, c1[1][1]);
    }

    const float bb0 = b_bilin[0], bb1 = b_bilin[1];
    #pragma unroll
    for (int ct = 0; ct < 2; ++ct) {
        const int j = j0 + ct * 16 + l16;
        const float dep0 = hd[j * 4 + 2], dep1 = hd[j * 4 + 3];
        #pragma unroll
        for (int rt = 0; rt < 2; ++rt) {
            #pragma unroll
            for (int r = 0; r < 8; ++r) {
                const int i = i0 + rt * 16 + r + 8 * half;
                const float v0 = c0[rt][ct][r] + bb0 + hd[i * 4 + 0] + dep0;
                const float v1 = c1[rt][ct][r] + bb1 + hd[i * 4 + 1] + dep1;
                Amat[(size_t)i * NN + j] = expf(v0) + expf(v1);
            }
        }
    }
}

// ---------------- per-node head/dep projections ----------------------
__global__ void headdep_kernel(const float* __restrict__ x,
                               const float* __restrict__ W_head,
                               const float* __restrict__ W_dep,
                               float* __restrict__ hd) {
    const int i = blockIdx.x * blockDim.x + threadIdx.x;
    if (i >= NN) return;
    const float* xi = x + (size_t)i * HH;
    float h0 = 0, h1 = 0, d0 = 0, d1 = 0;
    for (int t = 0; t < HH; ++t) {
        const float v = xi[t];
        h0 += v * W_head[t];  h1 += v * W_head[HH + t];
        d0 += v * W_dep[t];   d1 += v * W_dep[HH + t];
    }
    hd[i * 4 + 0] = h0; hd[i * 4 + 1] = h1;
    hd[i * 4 + 2] = d0; hd[i * 4 + 3] = d1;
}

// ---------------- root scores + exp (lap row 0) ----------------------
__global__ void root_kernel(const float* __restrict__ g,
                            const float* __restrict__ Wr2,
                            const float* __restrict__ br2,
                            float* __restrict__ rootScores,
                            float* __restrict__ expRoot) {
    const int i = blockIdx.x * blockDim.x + threadIdx.x;
    if (i >= NN) return;
    const float* gi = g + (size_t)i * HH;
    float s = br2[0];
    for (int t = 0; t < HH; ++t) s += gi[t] * Wr2[t];
    rootScores[i] = s;
    expRoot[i]    = expf(s);
}

// ---------------- gold: sparse edge sum of compat --------------------
__global__ void gold_edges_kernel(const float* __restrict__ xW0,
                                  const float* __restrict__ xW1,
                                  const float* __restrict__ x,
                                  const float* __restrict__ hd,
                                  const float* __restrict__ b_bilin,
                                  const int* __restrict__ parents,
                                  const int* __restrict__ children,
                                  float* __restrict__ gold_acc) {
    const int wave = blockIdx.x * (blockDim.x >> 5) + (threadIdx.x >> 5);
    const int lane = threadIdx.x & 31;
    if (wave >= EE) return;
    const int e = wave;
    const int p = parents[e], c = children[e];
    const int k = (e < EHALF) ? 0 : 1;
    const float* a = (k == 0 ? xW0 : xW1) + (size_t)p * HH;
    const float* b = x + (size_t)c * HH;
    float s = 0.f;
    for (int t = lane; t < HH; t += 32) s += a[t] * b[t];
    #pragma unroll
    for (int off = 16; off > 0; off >>= 1) s += __shfl_xor(s, off, 32);
    if (lane == 0) {
        s += b_bilin[k] + hd[p * 4 + k] + hd[c * 4 + 2 + k];
        atomicAdd(gold_acc, s);
    }
}

// ---------------- column sums of A -----------------------------------
__global__ void colsum_kernel(const float* __restrict__ Amat,
                              float* __restrict__ colsum) {
    const int j = blockIdx.x * blockDim.x + threadIdx.x;
    if (j >= NN) return;
    float s = 0.f;
    for (int i = 0; i < NN; ++i) s += Amat[(size_t)i * NN + j];
    colsum[j] = s;
}

// ---------------- lap build (in place over A) ------------------------
__global__ void build_lap_kernel(float* __restrict__ Amat,
                                 const float* __restrict__ colsum,
                                 const float* __restrict__ expRoot) {
    const size_t idx = (size_t)blockIdx.x * blockDim.x + threadIdx.x;
    if (idx >= (size_t)NN * NN) return;
    const int i = (int)(idx / NN), j = (int)(idx % NN);
    float v;
    if (i == 0)      v = expRoot[j];
    else {           v = -Amat[idx]; if (i == j) v += colsum[j]; }
    Amat[idx] = v;
}

// ---------------- LU: diagonal panel factor --------------------------
__global__ void lu_diag_kernel(float* __restrict__ M, int kb0,
                               float* __restrict__ logabs,
                               int* __restrict__ negcnt) {
    __shared__ float s[LUB][LUB + 1];
    const int t = threadIdx.x;                       // 0..63
    for (int j = 0; j < LUB; ++j)
        s[t][j] = M[(size_t)(kb0 + t) * NN + kb0 + j];
    __syncthreads();
    for (int k = 0; k < LUB - 1; ++k) {
        if (t > k) s[t][k] /= s[k][k];
        __syncthreads();
        if (t > k) {
            const float l = s[t][k];
            for (int j = k + 1; j < LUB; ++j) s[t][j] -= l * s[k][j];
        }
        __syncthreads();
    }
    for (int j = 0; j < LUB; ++j)
        M[(size_t)(kb0 + t) * NN + kb0 + j] = s[t][j];
    const float d = s[t][t];
    atomicAdd(logabs, logf(fabsf(d)));
    if (d < 0.f) atomicAdd(negcnt, 1);
}

// ---------------- LU: L21 = A21 * U11^{-1}  (thread-per-row) ---------
__global__ void lu_trsm_rows_kernel(float* __restrict__ M, int kb0,
                                    int r0, int R) {
    __shared__ float U[LUB][LUB + 1];
    for (int idx = threadIdx.x; idx < LUB * LUB; idx += blockDim.x) {
        const int i = idx / LUB, j = idx % LUB;
        U[i][j] = M[(size_t)(kb0 + i) * NN + kb0 + j];
    }
    __syncthreads();
    const int r = r0 + blockIdx.x * blockDim.x + threadIdx.x;
    if (r >= r0 + R) return;
    float xr[LUB];
    float* row = M + (size_t)r * NN + kb0;
    #pragma unroll
    for (int j = 0; j < LUB; ++j) {
        float v = row[j];
        #pragma unroll
        for (int t2 = 0; t2 < j; ++t2) v -= xr[t2] * U[t2][j];
        xr[j] = v / U[j][j];
    }
    #pragma unroll
    for (int j = 0; j < LUB; ++j) row[j] = xr[j];
}

// ---------------- LU: U12 = L11^{-1} * A12  (thread-per-column) ------
__global__ void lu_trsm_cols_kernel(float* __restrict__ M, int kb0,
                                    int c0, int Ccnt) {
    __shared__ float L[LUB][LUB + 1];
    for (int idx = threadIdx.x; idx < LUB * LUB; idx += blockDim.x) {
        const int i = idx / LUB, j = idx % LUB;
        L[i][j] = M[(size_t)(kb0 + i) * NN + kb0 + j];
    }
    __syncthreads();
    const int c = c0 + blockIdx.x * blockDim.x + threadIdx.x;
    if (c >= c0 + Ccnt) return;
    float xc[LUB];
    #pragma unroll
    for (int i = 0; i < LUB; ++i) {
        float v = M[(size_t)(kb0 + i) * NN + c];
        #pragma unroll
        for (int t2 = 0; t2 < i; ++t2) v -= L[i][t2] * xc[t2];
        xc[i] = v;
        M[(size_t)(kb0 + i) * NN + c] = v;
    }
}

// ---------------- LU: trailing A22 -= L21 @ U12  (WMMA 2x2) ----------
// Block = 4 waves; blockIdx.y selects a 32-row band of L21, staged into
// LDS (padded pitch kills bank conflicts); each wave does a 32x32 tile.
__global__ void lu_update_kernel(float* __restrict__ M, int kb0,
                                 int r0, int R) {
    __shared__ float As[LUB][33];                 // As[k][m], 32-row band
    const int tid = threadIdx.y * 32 + threadIdx.x;
    const int i0  = r0 + blockIdx.y * 32;
    for (int idx = tid; idx < 32 * LUB; idx += 128) {
        const int m = idx >> 6, k = idx & 63;     // coalesced 64-float runs
        As[k][m] = M[(size_t)(i0 + m) * NN + kb0 + k];
    }
    __syncthreads();
    const int bn32 = blockIdx.x * 4 + threadIdx.y;
    if (bn32 * 32 >= R) return;
    const int j0 = r0 + bn32 * 32;
    const int lane = threadIdx.x & 31, half = lane >> 4, l16 = lane & 15;

    v8f acc[2][2] = {};
    for (int k = 0; k < LUB; k += 4) {
        const int ka = k + half * 2;
        const float* Brow = M + (size_t)(kb0 + ka) * NN;
        v2f a0, a1, b0, b1;
        a0.x = As[ka][l16];        a0.y = As[ka + 1][l16];
        a1.x = As[ka][16 + l16];   a1.y = As[ka + 1][16 + l16];
        b0.x = Brow[j0 + l16];       b0.y = Brow[NN + j0 + l16];
        b1.x = Brow[j0 + 16 + l16];  b1.y = Brow[NN + j0 + 16 + l16];
        acc[0][0] = wmma4(a0, b0, acc[0][0]);
        acc[0][1] = wmma4(a0, b1, acc[0][1]);
        acc[1][0] = wmma4(a1, b0, acc[1][0]);
        acc[1][1] = wmma4(a1, b1, acc[1][1]);
    }
    #pragma unroll
    for (int rt = 0; rt < 2; ++rt)
        #pragma unroll
        for (int ct = 0; ct < 2; ++ct)
            #pragma unroll
            for (int r = 0; r < 8; ++r) {
                const int i = i0 + rt * 16 + r + 8 * half;
                const int j = j0 + ct * 16 + l16;
                M[(size_t)i * NN + j] -= acc[rt][ct][r];
            }
}

// ---------------- edge feature gather, TRANSPOSED [2H x EPAD] --------
__global__ void featsT_kernel(const float* __restrict__ x,
                              const int* __restrict__ parents,
                              const int* __restrict__ children,
                              float* __restrict__ featsT) {
    const size_t idx = (size_t)blockIdx.x * blockDim.x + threadIdx.x;
    if (idx >= (size_t)2 * HH * EPAD) return;
    const int t = (int)(idx / EPAD);
    const int e = (int)(idx % EPAD);
    float v = 0.f;
    if (e < EE) {
        const int node = (t < HH) ? parents[e] : children[e];
        const int d    = (t < HH) ? t : t - HH;
        v = x[(size_t)node * HH + d];
    }
    featsT[idx] = v;
}

// ---------------- 18-class head + CE sum -----------------------------
__global__ void rel_ce_kernel(const float* __restrict__ hrel,
                              const float* __restrict__ Wrel2,
                              const float* __restrict__ brel2,
                              const int* __restrict__ rels,
                              float* __restrict__ ce_acc) {
    const int e = blockIdx.x * blockDim.x + threadIdx.x;
    if (e >= EE) return;
    const float* h = hrel + (size_t)e * HH;
    float lg[NRELC];
    #pragma unroll
    for (int j = 0; j < NRELC; ++j) lg[j] = brel2[j];
    for (int d = 0; d < HH; ++d) {
        const float hv = h[d];
        const float* w = Wrel2 + (size_t)d * NRELC;
        #pragma unroll
        for (int j = 0; j < NRELC; ++j) lg[j] += hv * w[j];
    }
    float mx = lg[0];
    #pragma unroll
    for (int j = 1; j < NRELC; ++j) mx = fmaxf(mx, lg[j]);
    float se = 0.f;
    #pragma unroll
    for (int j = 0; j < NRELC; ++j) se += expf(lg[j] - mx);
    const float lse = mx + logf(se);
    const int cls = rels[e];
    atomicAdd(ce_acc, -(lg[cls] - lse));
}

// ---------------- scalar init / finalize -----------------------------
__global__ void init_kernel(float* scal, int* negcnt) {
    scal[0] = 0.f; scal[1] = 0.f; scal[2] = 0.f; negcnt[0] = 0;
}

__global__ void finalize_kernel(const float* __restrict__ scal,  // gold_e, logabs, ce
                                const int* __restrict__ negcnt,
                                const float* __restrict__ rootScores,
                                const int* __restrict__ true_root,
                                float* __restrict__ out) {
    const float gold    = scal[0] + rootScores[true_root[0]];
    const float logabsv = scal[1];
    const float ce      = scal[2];
    const float sign    = (negcnt[0] & 1) ? -1.f : 1.f;
    const float logdet  = (sign > 0.f) ? logabsv : nanf("");
    const float notnan  = (!isnan(gold) && !isnan(logdet)) ? 1.f : 0.f;
    const float mask    = (gold <= logdet * notnan) ? 1.f : 0.f;
    float loss          = (logdet - gold) * mask;
    if (isnan(loss)) loss = 0.f;
    out[0] = ALPHA * loss + ce;
}

// =====================================================================
extern "C" void kernel_launch(void* const* d_in, const int* in_sizes, int n_in,
                              void* d_out, int out_size, void* d_ws, size_t ws_size,
                              hipStream_t stream) {
    const float* x       = (const float*)d_in[0];   // all_edus[0]  [N,H]
    const float* W_bilin = (const float*)d_in[3];   // [2,H,H]
    const float* b_bilin = (const float*)d_in[4];
    const float* W_head  = (const float*)d_in[5];
    const float* W_dep   = (const float*)d_in[6];
    const float* Wr1     = (const float*)d_in[7];
    const float* br1     = (const float*)d_in[8];
    const float* Wr2     = (const float*)d_in[9];
    const float* br2     = (const float*)d_in[10];
    const float* Wrel1   = (const float*)d_in[11];
    const float* brel1   = (const float*)d_in[12];
    const float* Wrel2   = (const float*)d_in[13];
    const float* brel2   = (const float*)d_in[14];
    const int* children  = (const int*)d_in[15];
    const int* parents   = (const int*)d_in[16];
    const int* rels      = (const int*)d_in[17];
    const int* true_root = (const int*)d_in[18];
    float* out = (float*)d_out;

    // -------- workspace layout (floats) --------
    float* ws = (float*)d_ws;
    size_t o = 0;
    float* xW0        = ws + o; o += (size_t)NN * HH;   // row-major (edge gather)
    float* xW1        = ws + o; o += (size_t)NN * HH;
    float* xW0T       = ws + o; o += (size_t)NN * HH;   // [H x N] (compat A)
    float* xW1T       = ws + o; o += (size_t)NN * HH;
    float* xT         = ws + o; o += (size_t)NN * HH;   // [H x N]
    float* g          = ws + o; o += (size_t)NN * HH;
    float* hrel       = ws + o; o += (size_t)EPAD * HH;
    float* featsT     = ws + o; o += (size_t)EPAD * 2 * HH;  // [2H x EPAD]
    float* hd         = ws + o; o += (size_t)NN * 4;
    float* rootScores = ws + o; o += NN;
    float* expRoot    = ws + o; o += NN;
    float* colsum     = ws + o; o += NN;
    float* scal       = ws + o; o += 4;     // [gold_edges, logabs, ce, pad]
    int*   negcnt     = (int*)(ws + o); o += 4;
    float* Amat       = ws + o; o += (size_t)NN * NN;   // A -> lap -> LU, in place

    const dim3 wblk(32, 8);     // 8 waves / block for 1-tile WMMA kernels
    const dim3 wblk4(32, 4);    // 4 waves / block for 2x2-tile WMMA kernels

    init_kernel<<<1, 1, 0, stream>>>(scal, negcnt);

    // xT = x^T  (so all downstream A-fragments load coalesced)
    transpose_kernel<<<dim3(HH / 32, NN / 32), dim3(32, 8), 0, stream>>>(x, xT, NN, HH);

    // projections (WMMA f32, AT-layout A side)
    {
        const int tiles = (NN / 16) * (HH / 16);      // 3072
        gemm_at_dual_kernel<<<tiles / 8, wblk, 0, stream>>>(xT, W_bilin,           xW0, xW0T, NN, HH, HH, NN);
        gemm_at_dual_kernel<<<tiles / 8, wblk, 0, stream>>>(xT, W_bilin + HH * HH, xW1, xW1T, NN, HH, HH, NN);
        gemm_at_bias_gelu_kernel<<<tiles / 8, wblk, 0, stream>>>(xT, Wr1, br1, g, NN, HH, HH, NN);
    }

    headdep_kernel<<<(NN + 255) / 256, 256, 0, stream>>>(x, W_head, W_dep, hd);
    root_kernel<<<(NN + 255) / 256, 256, 0, stream>>>(g, Wr2, br2, rootScores, expRoot);

    // A = sum_k exp(compat_k): fused dual N x N x H WMMA GEMM, 32x32/wave
    {
        const int waves = (NN / 32) * (NN / 32);      // 9216
        compat_exp_kernel<<<waves / 4, wblk4, 0, stream>>>(xW0T, xW1T, xT, hd, b_bilin, Amat);
    }

    // gold edge sum via sparse edges (skips both dense adjacency matrices)
    gold_edges_kernel<<<(EE + 7) / 8, 256, 0, stream>>>(xW0, xW1, x, hd, b_bilin,
                                                        parents, children, scal + 0);

    colsum_kernel<<<(NN + 255) / 256, 256, 0, stream>>>(Amat, colsum);
    build_lap_kernel<<<(int)(((size_t)NN * NN + 255) / 256), 256, 0, stream>>>(Amat, colsum, expRoot);

    // blocked LU (no pivoting; sign tracked) — WMMA trailing updates
    for (int kb = 0; kb < NN / LUB; ++kb) {
        const int kb0 = kb * LUB;
        const int r0  = kb0 + LUB;
        const int R   = NN - r0;
        lu_diag_kernel<<<1, LUB, 0, stream>>>(Amat, kb0, scal + 1, negcnt);
        if (R > 0) {
            lu_trsm_rows_kernel<<<(R + 255) / 256, 256, 0, stream>>>(Amat, kb0, r0, R);
            lu_trsm_cols_kernel<<<(R + 255) / 256, 256, 0, stream>>>(Amat, kb0, r0, R);
            const int bands = R / 32;
            lu_update_kernel<<<dim3((bands + 3) / 4, bands), wblk4, 0, stream>>>(Amat, kb0, r0, R);
        }
    }

    // relation MLP: transposed gather -> WMMA GEMM + gelu -> CE
    {
        const size_t total = (size_t)2 * HH * EPAD;
        featsT_kernel<<<(int)((total + 255) / 256), 256, 0, stream>>>(x, parents, children, featsT);
        const int tiles = (EPAD / 16) * (HH / 16);    // 3072
        gemm_at_bias_gelu_kernel<<<tiles / 8, wblk, 0, stream>>>(featsT, Wrel1, brel1, hrel,
                                                                 EPAD, HH, 2 * HH, EPAD);
        rel_ce_kernel<<<(EE + 255) / 256, 256, 0, stream>>>(hrel, Wrel2, brel2, rels, scal + 2);
    }

    finalize_kernel<<<1, 1, 0, stream>>>(scal, negcnt, rootScores, true_root, out);
}